// Integrator_26276609917759
// MI455X (gfx1250) — compile-verified
//
#include <hip/hip_runtime.h>
#include <cstdint>

typedef float v4f __attribute__((ext_vector_type(4)));

#define VOLD 256
#define NVOX ((size_t)(256*256*256))
#define MAXW 500.0f

__device__ __forceinline__ bool volidx(int x, int y, int z, int& lin) {
    bool v = ((unsigned)x < 256u) & ((unsigned)y < 256u) & ((unsigned)z < 256u);
    lin = (x << 16) + (y << 8) + z;   // ys*zs = 65536, zs = 256
    return v;
}

// ---------------------------------------------------------------------------
// Pass 1: scatter regular + empty updates.
//   out[0      .. N)   : sum_upd accumulator (later v_new)
//   out[9N     ..10N)  : sum_w accumulator   (later w_new)
//   out[11N    ..12N)  : sum_we accumulator  (later occupancy mask)
// touched flags live in d_ws as bytes (plain stores of 1 — benign races).
// ---------------------------------------------------------------------------
__global__ __launch_bounds__(256) void scatter_vw(
    const float* __restrict__ vals, const int* __restrict__ idx,
    const float* __restrict__ w,    const int* __restrict__ idxe,
    const float* __restrict__ we,   float* __restrict__ out,
    unsigned char* __restrict__ tch, unsigned char* __restrict__ tche, int n)
{
    int tid = blockIdx.x * 256 + threadIdx.x;
    if (tid >= n) return;
    int b = 3 * tid, lin;
    if (volidx(idx[b], idx[b + 1], idx[b + 2], lin)) {
        float wi = w[tid];
        float v  = vals[tid >> 3];          // each point value feeds 8 corners
        atomicAdd(out + lin, wi * v);
        atomicAdd(out + 9 * NVOX + lin, wi);
        tch[lin] = 1;
    }
    if (volidx(idxe[b], idxe[b + 1], idxe[b + 2], lin)) {
        atomicAdd(out + 11 * NVOX + lin, we[tid]);
        tche[lin] = 1;
    }
}

// ---------------------------------------------------------------------------
// Pass 2: scatter 8-wide features.
//   out[N  .. 9N) : sum_feat accumulator (later feat_new), row-major (vox,8)
//   out[10N..11N) : cnt_feat accumulator (later fw_new)
// ---------------------------------------------------------------------------
__global__ __launch_bounds__(256) void scatter_feat(
    const float* __restrict__ feats, const int* __restrict__ fidx,
    float* __restrict__ out, int npts)
{
    int tid = blockIdx.x * 256 + threadIdx.x;
    if (tid >= npts) return;
    int b = 3 * tid, lin;
    if (!volidx(fidx[b], fidx[b + 1], fidx[b + 2], lin)) return;
    const v4f* fp = (const v4f*)(feats + (size_t)tid * 8);
    v4f a = fp[0], c = fp[1];
    float* dst = out + NVOX + (size_t)lin * 8;
    atomicAdd(dst + 0, a.x); atomicAdd(dst + 1, a.y);
    atomicAdd(dst + 2, a.z); atomicAdd(dst + 3, a.w);
    atomicAdd(dst + 4, c.x); atomicAdd(dst + 5, c.y);
    atomicAdd(dst + 6, c.z); atomicAdd(dst + 7, c.w);
    atomicAdd(out + 10 * NVOX + lin, 1.0f);
}

// ---------------------------------------------------------------------------
// Pass 3: dense per-voxel normalization (HBM-bound streaming).
// CDNA5 async global->LDS copy stages the 32B/lane feature rows; non-temporal
// loads/stores keep the once-streamed 800MB out of L2.
// ---------------------------------------------------------------------------
__global__ __launch_bounds__(256) void finalize(
    const float* __restrict__ vvol, const float* __restrict__ wvol,
    const float* __restrict__ fvol, const float* __restrict__ fwvol,
    const unsigned char* __restrict__ tch, const unsigned char* __restrict__ tche,
    float* __restrict__ out)
{
    const size_t tid = (size_t)blockIdx.x * 256 + threadIdx.x;
    __shared__ float ldsF[256 * 8];

    {   // async-copy this block's feature rows into LDS (tracked by ASYNCcnt)
        unsigned long long ga = (unsigned long long)(const void*)(fvol + tid * 8);
        unsigned int la = (unsigned int)(uintptr_t)&ldsF[threadIdx.x * 8];
        asm volatile("global_load_async_to_lds_b128 %0, %2, off\n\t"
                     "global_load_async_to_lds_b128 %1, %3, off"
                     :: "v"(la), "v"(la + 16u), "v"(ga), "v"(ga + 16ull)
                     : "memory");
    }

    float vv  = __builtin_nontemporal_load(vvol  + tid);
    float wv  = __builtin_nontemporal_load(wvol  + tid);
    float fw  = __builtin_nontemporal_load(fwvol + tid);
    float su  = out[tid];
    float sw  = out[ 9 * NVOX + tid];
    float cf  = out[10 * NVOX + tid];
    float swe = out[11 * NVOX + tid];
    bool  t   = tch[tid]  != 0;
    bool  te  = tche[tid] != 0;
    const v4f* sfp = (const v4f*)(out + NVOX + tid * 8);
    v4f sf0 = sfp[0], sf1 = sfp[1];

    // value / weight volumes
    float wvvv = wv * vv;
    float vu   = (wvvv + su)          / (wv + sw);
    float vue  = (wvvv + 0.1f * swe)  / (wv + swe);
    float vn   = te ? vue : (t ? vu : vv);
    float wu   = fminf(fmaxf(wv + sw , 0.0f), MAXW);
    float wue  = fminf(fmaxf(wv + swe, 0.0f), MAXW);
    float wn   = te ? wue : (t ? wu : wv);

    asm volatile("s_wait_asynccnt 0" ::: "memory");
    v4f fa = *(const v4f*)&ldsF[threadIdx.x * 8];
    v4f fb = *(const v4f*)&ldsF[threadIdx.x * 8 + 4];

    // feature volume
    bool  ft = cf > 0.0f;
    float fd = fw + cf;
    v4f  f0  = ft ? (fw * fa + sf0) / fd : fa;
    v4f  f1  = ft ? (fw * fb + sf1) / fd : fb;
    float fwn = ft ? fminf(fmaxf(fd, 0.0f), MAXW) : fw;

    __builtin_nontemporal_store(vn,   out + tid);
    __builtin_nontemporal_store(wn,   out +  9 * NVOX + tid);
    __builtin_nontemporal_store(fwn,  out + 10 * NVOX + tid);
    __builtin_nontemporal_store(0.0f, out + 11 * NVOX + tid);  // clear for mask pass
    v4f* fo = (v4f*)(out + NVOX + tid * 8);
    __builtin_nontemporal_store(f0, fo);
    __builtin_nontemporal_store(f1, fo + 1);
}

// ---------------------------------------------------------------------------
// Pass 4: occupancy mask from filter_indices (runs after finalize cleared it).
// ---------------------------------------------------------------------------
__global__ __launch_bounds__(256) void scatter_filter(
    const int* __restrict__ fil, float* __restrict__ mask, int n)
{
    int tid = blockIdx.x * 256 + threadIdx.x;
    if (tid >= n) return;
    int b = 3 * tid, lin;
    if (volidx(fil[b], fil[b + 1], fil[b + 2], lin))
        mask[lin] = 1.0f;   // all writers store 1.0f — race benign
}

extern "C" void kernel_launch(void* const* d_in, const int* in_sizes, int n_in,
                              void* d_out, int out_size, void* d_ws, size_t ws_size,
                              hipStream_t stream) {
    const float* upd_vals = (const float*)d_in[0];   // (H,W)
    const float* upd_feat = (const float*)d_in[1];   // (H,W,1,8)
    const int*   upd_idx  = (const int*)d_in[2];     // (H,W,8,3)
    const int*   upd_fidx = (const int*)d_in[3];     // (H,W,1,3)
    const int*   fil_idx  = (const int*)d_in[4];     // (H,W,8,3)
    const float* upd_w    = (const float*)d_in[5];   // (H,W,8)
    const int*   idx_e    = (const int*)d_in[6];     // (H,W,8,3)
    const float* w_e      = (const float*)d_in[7];   // (H,W,8)
    const float* vvol     = (const float*)d_in[8];   // 256^3
    const float* fvol     = (const float*)d_in[9];   // 256^3 x 8
    const float* wvol     = (const float*)d_in[10];  // 256^3
    const float* fwvol    = (const float*)d_in[11];  // 256^3

    const size_t N = (size_t)in_sizes[8];            // 16,777,216
    float* out = (float*)d_out;
    unsigned char* tch  = (unsigned char*)d_ws;
    unsigned char* tche = (unsigned char*)d_ws + N;

    // zero all accumulators (entire 12N-float output doubles as accumulators)
    hipMemsetAsync(d_out, 0, 12 * N * sizeof(float), stream);
    hipMemsetAsync(d_ws,  0, 2 * N, stream);

    int nupd = in_sizes[5];                          // H*W*8
    int npts = in_sizes[0];                          // H*W
    scatter_vw  <<<(nupd + 255) / 256, 256, 0, stream>>>(upd_vals, upd_idx, upd_w,
                                                         idx_e, w_e, out, tch, tche, nupd);
    scatter_feat<<<(npts + 255) / 256, 256, 0, stream>>>(upd_feat, upd_fidx, out, npts);
    finalize    <<<(unsigned)(N / 256), 256, 0, stream>>>(vvol, wvol, fvol, fwvol,
                                                          tch, tche, out);
    scatter_filter<<<(nupd + 255) / 256, 256, 0, stream>>>(fil_idx, out + 11 * NVOX, nupd);
}